// TransformerOCR_17111149707870
// MI455X (gfx1250) — compile-verified
//
#include <hip/hip_runtime.h>
#include <hip/hip_bf16.h>
#include <math.h>

typedef __attribute__((ext_vector_type(2))) float v2f;
typedef __attribute__((ext_vector_type(8))) float v8f;

#define E_DIM   1024
#define N_TOK   128
#define H_NUM   16
#define HD      64
#define S_TOT   512
#define S_PREV  511

// ---------------------------------------------------------------------------
// FP32 GEMM tile via V_WMMA_F32_16X16X4_F32.
// A: [M,K] row-major (activations), B == W^T where W: [N,K] row-major.
// A-tile 16x4 layout: lanes 0-15 hold row m, K={k,k+1}; lanes 16-31 K={k+2,k+3}
//   -> one float2 load per lane.
// B-tile 4x16 layout mirrors it: lanes 0-15 hold col n at K={k,k+1}, i.e.
//   W[n][k..k+1] -> also one contiguous float2 load per lane.
// C/D 16x16 f32: VGPR i -> M = m0 + i (+8 for lanes 16-31), N = n0 + (lane&15)
// ---------------------------------------------------------------------------
__device__ __forceinline__ v8f gemm_tile_f32(const float* __restrict__ A,
                                             const float* __restrict__ W,
                                             int m0, int n0, int K, int lane) {
    const int r  = lane & 15;
    const int kh = (lane >> 4) << 1;          // 0 for lanes 0-15, 2 for 16-31
    const float* ap = A + (size_t)(m0 + r) * K + kh;
    const float* bp = W + (size_t)(n0 + r) * K + kh;
    v8f c = {};
    for (int k = 0; k < K; k += 4) {
        v2f a = *(const v2f*)(ap + k);
        v2f b = *(const v2f*)(bp + k);
        c = __builtin_amdgcn_wmma_f32_16x16x4_f32(false, a, false, b,
                                                  (short)0, c, false, false);
    }
    return c;
}

// QKV projection: qkv[m, j] = query[m,:] . W[j,:] + bias[j], j in [0,3072)
// Routed: j<E -> q (scaled by 0.125), j<2E -> k_new, else v_new.
__global__ __launch_bounds__(128) void qkv_proj_kernel(
    const float* __restrict__ query, const float* __restrict__ W,
    const float* __restrict__ bias,
    float* __restrict__ q_s, float* __restrict__ k_new, float* __restrict__ v_new)
{
    const int wave = threadIdx.x >> 5;
    const int lane = threadIdx.x & 31;
    const int m0 = blockIdx.x * 16;
    const int n0 = (blockIdx.y * 4 + wave) * 16;

    v8f c = gemm_tile_f32(query, W, m0, n0, E_DIM, lane);

    const int ncol = n0 + (lane & 15);
    const int mb   = m0 + ((lane >> 4) << 3);
    const float bv = bias[ncol];
    if (ncol < E_DIM) {                         // q head (uniform per wave)
        #pragma unroll
        for (int i = 0; i < 8; ++i)
            q_s[(size_t)(mb + i) * E_DIM + ncol] = (c[i] + bv) * 0.125f;
    } else if (ncol < 2 * E_DIM) {              // k head
        #pragma unroll
        for (int i = 0; i < 8; ++i)
            k_new[(size_t)(mb + i) * E_DIM + (ncol - E_DIM)] = c[i] + bv;
    } else {                                    // v head
        #pragma unroll
        for (int i = 0; i < 8; ++i)
            v_new[(size_t)(mb + i) * E_DIM + (ncol - 2 * E_DIM)] = c[i] + bv;
    }
}

// Attention core: one block of 8 wave32s per (n,h); streams 512 K rows + 512 V
// rows (256B each) with non-temporal loads. Two-pass softmax via LDS scores.
__global__ __launch_bounds__(256) void attn_kernel(
    const float* __restrict__ q_s, const float* __restrict__ prev_kv,
    const float* __restrict__ k_new, const float* __restrict__ v_new,
    float* __restrict__ attn_out)
{
    __shared__ float sc[S_TOT];
    __shared__ float part[8][HD];
    __shared__ float redbuf[8];
    __shared__ float gshared[2];

    const int b    = blockIdx.x;     // n*16 + h
    const int n    = b >> 4;
    const int h    = b & 15;
    const int tid  = threadIdx.x;
    const int wave = tid >> 5;
    const int lane = tid & 31;
    const int d0   = lane << 1;      // 2 fp32 elements of the 64-wide head/lane

    const size_t head_off = (size_t)h * HD + d0;
    const float* qp = q_s + (size_t)n * E_DIM + head_off;
    const float qx = qp[0], qy = qp[1];

    // ---- pass 1: scores ----------------------------------------------------
    const int sbeg = wave * (S_TOT / 8);
    for (int s = sbeg; s < sbeg + (S_TOT / 8); ++s) {
        const float* kp = (s < S_PREV)
            ? prev_kv + ((size_t)s * N_TOK + n) * (2 * E_DIM) + head_off
            : k_new + (size_t)n * E_DIM + head_off;
        v2f kv = __builtin_nontemporal_load((const v2f*)kp);
        float p = qx * kv.x + qy * kv.y;
        #pragma unroll
        for (int off = 16; off; off >>= 1) p += __shfl_xor(p, off, 32);
        if (lane == 0) sc[s] = p;
    }
    __syncthreads();

    // ---- softmax: global max -----------------------------------------------
    float m = fmaxf(sc[tid], sc[tid + 256]);
    #pragma unroll
    for (int off = 16; off; off >>= 1) m = fmaxf(m, __shfl_xor(m, off, 32));
    if (lane == 0) redbuf[wave] = m;
    __syncthreads();
    if (tid == 0) {
        float mm = redbuf[0];
        #pragma unroll
        for (int i = 1; i < 8; ++i) mm = fmaxf(mm, redbuf[i]);
        gshared[0] = mm;
    }
    __syncthreads();
    const float gmax = gshared[0];

    // ---- exponentiate in place + global sum --------------------------------
    float e0 = __expf(sc[tid] - gmax);
    float e1 = __expf(sc[tid + 256] - gmax);
    sc[tid] = e0; sc[tid + 256] = e1;           // each thread owns its 2 slots
    float ssum = e0 + e1;
    #pragma unroll
    for (int off = 16; off; off >>= 1) ssum += __shfl_xor(ssum, off, 32);
    if (lane == 0) redbuf[wave] = ssum;
    __syncthreads();
    if (tid == 0) {
        float ss = 0.f;
        #pragma unroll
        for (int i = 0; i < 8; ++i) ss += redbuf[i];
        gshared[1] = ss;
    }
    __syncthreads();

    // ---- pass 2: weighted V accumulation -----------------------------------
    float ax = 0.f, ay = 0.f;
    for (int s = sbeg; s < sbeg + (S_TOT / 8); ++s) {
        const float* vp = (s < S_PREV)
            ? prev_kv + ((size_t)s * N_TOK + n) * (2 * E_DIM) + E_DIM + head_off
            : v_new + (size_t)n * E_DIM + head_off;
        v2f vv = __builtin_nontemporal_load((const v2f*)vp);
        const float w = sc[s];
        ax += w * vv.x; ay += w * vv.y;
    }
    part[wave][d0]     = ax;
    part[wave][d0 + 1] = ay;
    __syncthreads();

    const float inv = 1.0f / gshared[1];
    if (tid < HD) {
        float t = 0.f;
        #pragma unroll
        for (int i = 0; i < 8; ++i) t += part[i][tid];
        attn_out[(size_t)n * E_DIM + (size_t)h * HD + tid] = t * inv;
    }
}

// Output projection: out[m, j] = attn[m,:] . Wo[j,:] + bias[j]
__global__ __launch_bounds__(128) void out_proj_kernel(
    const float* __restrict__ A, const float* __restrict__ W,
    const float* __restrict__ bias, float* __restrict__ out)
{
    const int wave = threadIdx.x >> 5;
    const int lane = threadIdx.x & 31;
    const int m0 = blockIdx.x * 16;
    const int n0 = (blockIdx.y * 4 + wave) * 16;

    v8f c = gemm_tile_f32(A, W, m0, n0, E_DIM, lane);

    const int ncol = n0 + (lane & 15);
    const int mb   = m0 + ((lane >> 4) << 3);
    const float bv = bias[ncol];
    #pragma unroll
    for (int i = 0; i < 8; ++i)
        out[(size_t)(mb + i) * E_DIM + ncol] = c[i] + bv;
}

extern "C" void kernel_launch(void* const* d_in, const int* in_sizes, int n_in,
                              void* d_out, int out_size, void* d_ws, size_t ws_size,
                              hipStream_t stream) {
    (void)in_sizes; (void)n_in; (void)out_size; (void)ws_size;

    const float* query   = (const float*)d_in[0];   // [1,128,1024]
    const float* prev_kv = (const float*)d_in[1];   // [511,128,2048]
    const float* ipw     = (const float*)d_in[2];   // [3072,1024]
    const float* ipb     = (const float*)d_in[3];   // [3072]
    const float* opw     = (const float*)d_in[4];   // [1024,1024]
    const float* opb     = (const float*)d_in[5];   // [1024]
    float* out = (float*)d_out;                     // [1,128,1024]

    float* ws    = (float*)d_ws;
    float* q_s   = ws;                              // [128,1024] pre-scaled q
    float* k_new = ws + 1 * N_TOK * E_DIM;          // [128,1024]
    float* v_new = ws + 2 * N_TOK * E_DIM;          // [128,1024]
    float* attn  = ws + 3 * N_TOK * E_DIM;          // [128,1024]

    // QKV projection: M=128 (8 tiles), N=3072 (192 tiles, 4 waves/block)
    qkv_proj_kernel<<<dim3(8, 48), 128, 0, stream>>>(query, ipw, ipb,
                                                     q_s, k_new, v_new);
    // Attention: one block per (n, h) pair
    attn_kernel<<<dim3(N_TOK * H_NUM), 256, 0, stream>>>(q_s, prev_kv,
                                                         k_new, v_new, attn);
    // Output projection: M=128 (8 tiles), N=1024 (64 tiles, 4 waves/block)
    out_proj_kernel<<<dim3(8, 16), 128, 0, stream>>>(attn, opw, opb, out);
}